// GCNGAE_75969381531758
// MI455X (gfx1250) — compile-verified
//
#include <hip/hip_runtime.h>
#include <hip/hip_bf16.h>

typedef __attribute__((ext_vector_type(2))) float v2f;
typedef __attribute__((ext_vector_type(8))) float v8f;

// ---------------------------------------------------------------------------
// init: deg=1 (self loop), zero layer-1 accumulator and final output
// ---------------------------------------------------------------------------
__global__ void k_init(unsigned* __restrict__ deg, float* __restrict__ agg1,
                       float* __restrict__ out, int n, int n64, int n32) {
    int i = blockIdx.x * blockDim.x + threadIdx.x;
    if (i < n)   deg[i]  = 1u;
    if (i < n64) agg1[i] = 0.0f;
    if (i < n32) out[i]  = 0.0f;
}

// deg[col[e]] += 1
__global__ void k_deg(const long long* __restrict__ col, unsigned* __restrict__ deg, int E) {
    int e = blockIdx.x * blockDim.x + threadIdx.x;
    if (e < E) atomicAdd(&deg[(int)col[e]], 1u);
}

// dinv[i] = deg[i]^-1/2  (deg >= 1 always, self-loop)
__global__ void k_dinv(const unsigned* __restrict__ deg, float* __restrict__ dinv, int n) {
    int i = blockIdx.x * blockDim.x + threadIdx.x;
    if (i < n) dinv[i] = rsqrtf((float)deg[i]);
}

// ---------------------------------------------------------------------------
// Dense fp32 GEMM: Y[N,F] = X[N,K] * W[K,F] via V_WMMA_F32_16X16X4_F32.
// One wave per 16x16 output tile; N assumed multiple of 16 (100000 = 6250*16).
// A 16x4 layout: lanes 0-15 hold K={0,1}, lanes 16-31 hold K={2,3} (M = lane%16).
// B 4x16 layout: VGPR0 rows K={0|2}, VGPR1 rows K={1|3} striped across lanes (N = lane%16).
// C/D 16x16: VGPR v = row v (lanes 0-15) / row v+8 (lanes 16-31).
// ---------------------------------------------------------------------------
template <int K, int F>
__global__ void k_gemm_wmma(const float* __restrict__ X, const float* __restrict__ W,
                            float* __restrict__ Y, int nTiles) {
    int wave = (blockIdx.x * blockDim.x + threadIdx.x) >> 5;
    if (wave >= nTiles) return;                     // wave-uniform: EXEC stays all-1 for WMMA
    constexpr int CT = F / 16;
    int row0 = (wave / CT) * 16;
    int col0 = (wave % CT) * 16;
    int lane = threadIdx.x & 31;
    int half = lane >> 4;                           // 0: lanes 0-15, 1: lanes 16-31
    int mn   = lane & 15;

    v8f c = {};
    const float* xrow = X + (long long)(row0 + mn) * K;
    const float* wcol = W + col0 + mn;
    for (int k = 0; k < K; k += 4) {
        v2f a, b;
        a.x = xrow[k + 2 * half + 0];
        a.y = xrow[k + 2 * half + 1];
        b.x = wcol[(long long)(k + 2 * half + 0) * F];
        b.y = wcol[(long long)(k + 2 * half + 1) * F];
        c = __builtin_amdgcn_wmma_f32_16x16x4_f32(false, a, false, b,
                                                  (short)0, c, false, false);
    }
    float* yrow = Y + col0 + mn;
    #pragma unroll
    for (int v = 0; v < 8; ++v) {
        int m = v + 8 * half;
        yrow[(long long)(row0 + m) * F] = c[v];
    }
}

// ---------------------------------------------------------------------------
// Edge aggregation: out[col] += t[row] * dinv[row]*dinv[col].
// One thread per (edge, 4-float chunk); chunk-major within an edge so a wave
// reads 2 edges' messages fully coalesced. Items e >= E are self-loops.
// fp32 atomics lower to global_atomic_add_f32 (L2-resident RMW).
// ---------------------------------------------------------------------------
template <int F>
__global__ void k_scatter(const float* __restrict__ t,
                          const long long* __restrict__ row,
                          const long long* __restrict__ col,
                          const float* __restrict__ dinv,
                          float* __restrict__ out, int E, int N) {
    constexpr int CH = F / 4;
    int tid = blockIdx.x * blockDim.x + threadIdx.x;
    int total = (E + N) * CH;
    if (tid >= total) return;
    int e  = tid / CH;
    int ch = tid % CH;
    int r, c;
    if (e < E) { r = (int)row[e]; c = (int)col[e]; }
    else       { r = c = e - E; }
    float w = dinv[r] * dinv[c];
    const float4 v = *(const float4*)(t + (long long)r * F + ch * 4);
    float* o = out + (long long)c * F + ch * 4;
    unsafeAtomicAdd(o + 0, v.x * w);
    unsafeAtomicAdd(o + 1, v.y * w);
    unsafeAtomicAdd(o + 2, v.z * w);
    unsafeAtomicAdd(o + 3, v.w * w);
}

// h = relu(agg + b)  (in place)
__global__ void k_bias_relu(float* __restrict__ h, const float* __restrict__ b, int total, int F) {
    int i = blockIdx.x * blockDim.x + threadIdx.x;
    if (i < total) h[i] = fmaxf(h[i] + b[i % F], 0.0f);
}

// out += b
__global__ void k_bias(float* __restrict__ o, const float* __restrict__ b, int total, int F) {
    int i = blockIdx.x * blockDim.x + threadIdx.x;
    if (i < total) o[i] = o[i] + b[i % F];
}

// ---------------------------------------------------------------------------
extern "C" void kernel_launch(void* const* d_in, const int* in_sizes, int n_in,
                              void* d_out, int out_size, void* d_ws, size_t ws_size,
                              hipStream_t stream) {
    constexpr int IN = 128, HID = 64, OUT = 32;
    const float*     x  = (const float*)d_in[0];
    const long long* ei = (const long long*)d_in[1];   // int64 [2, E]
    const float*     W1 = (const float*)d_in[2];
    const float*     b1 = (const float*)d_in[3];
    const float*     W2 = (const float*)d_in[4];
    const float*     b2 = (const float*)d_in[5];
    float* out = (float*)d_out;

    const int N = in_sizes[0] / IN;                    // 100000
    const int E = in_sizes[1] / 2;                     // 1600000
    const long long* row = ei;
    const long long* col = ei + E;

    // workspace layout (256B aligned)
    auto align = [](size_t v) { return (v + 255) & ~(size_t)255; };
    char* ws = (char*)d_ws;
    size_t off = 0;
    unsigned* deg  = (unsigned*)(ws + off); off += align((size_t)N * 4);
    float*    dinv = (float*)   (ws + off); off += align((size_t)N * 4);
    float*    t1   = (float*)   (ws + off); off += align((size_t)N * HID * 4);
    float*    h    = (float*)   (ws + off); off += align((size_t)N * HID * 4);
    float*    t2   = (float*)   (ws + off); off += align((size_t)N * OUT * 4);
    (void)ws_size;

    const int B = 256;
    const int n64 = N * HID, n32 = N * OUT;

    // 1) init deg=1, agg1=0, out=0
    k_init<<<(n64 + B - 1) / B, B, 0, stream>>>(deg, h, out, N, n64, n32);
    // 2) degree count over col
    k_deg<<<(E + B - 1) / B, B, 0, stream>>>(col, deg, E);
    // 3) dinv = rsqrt(deg)
    k_dinv<<<(N + B - 1) / B, B, 0, stream>>>(deg, dinv, N);
    // 4) t1 = x @ W1   (fp32 WMMA)
    {
        int tiles = (N / 16) * (HID / 16);
        int blocks = (tiles + 7) / 8;                  // 8 waves / 256-thread block
        k_gemm_wmma<IN, HID><<<blocks, B, 0, stream>>>(x, W1, t1, tiles);
    }
    // 5) agg1 (in h buffer) += norm-weighted gather/scatter of t1
    {
        int total = (E + N) * (HID / 4);
        k_scatter<HID><<<(total + B - 1) / B, B, 0, stream>>>(t1, row, col, dinv, h, E, N);
    }
    // 6) h = relu(agg1 + b1)
    k_bias_relu<<<(n64 + B - 1) / B, B, 0, stream>>>(h, b1, n64, HID);
    // 7) t2 = h @ W2   (fp32 WMMA)
    {
        int tiles = (N / 16) * (OUT / 16);
        int blocks = (tiles + 7) / 8;
        k_gemm_wmma<HID, OUT><<<blocks, B, 0, stream>>>(h, W2, t2, tiles);
    }
    // 8) out += norm-weighted gather/scatter of t2
    {
        int total = (E + N) * (OUT / 4);
        k_scatter<OUT><<<(total + B - 1) / B, B, 0, stream>>>(t2, row, col, dinv, out, E, N);
    }
    // 9) out += b2
    k_bias<<<(n32 + B - 1) / B, B, 0, stream>>>(out, b2, n32, OUT);
}